// masking_threshold_loss_12378095747804
// MI455X (gfx1250) — compile-verified
//
#include <hip/hip_runtime.h>
#include <hip/hip_bf16.h>
#include <stdint.h>

#define M_PIX     65536
#define AT_LEAST_I 10u
#define THRESH    3e-05f

typedef __attribute__((ext_vector_type(2))) float v2f;
typedef __attribute__((ext_vector_type(8))) float v8f;

// Map float -> u32 key whose ascending order == descending float order.
__device__ __forceinline__ uint32_t desc_key(float x) {
    uint32_t b = __float_as_uint(x);
    uint32_t u = (b & 0x80000000u) ? ~b : (b | 0x80000000u); // ascending-float key
    return ~u;                                               // flip for descending
}

__global__ void zero_S(float* __restrict__ S) {
    int i = blockIdx.x * blockDim.x + threadIdx.x;
    if (i < M_PIX) S[i] = 0.f;
}

// One workgroup (1024 threads = 32 waves) per row. 16-bit-radix counting sort
// with the full 65536-bin histogram resident in CDNA5's 320KB LDS.
__global__ void rank_deposit(const float* __restrict__ x, const float* __restrict__ p,
                             uint32_t* __restrict__ rec, float* __restrict__ S) {
    extern __shared__ uint32_t lds[];
    uint32_t* hist  = lds;               // 65536 u32 = 256 KB
    uint32_t* wtot  = lds + 65536;       // 32 u32
    uint32_t* wbase = lds + 65536 + 32;  // 32 u32
    const int t    = threadIdx.x;
    const int lane = t & 31;
    const int w    = t >> 5;
    const int row  = blockIdx.x;
    const float*   xr   = x   + (size_t)row * M_PIX;
    const float*   pr   = p   + (size_t)row * M_PIX;
    uint32_t*      recr = rec + (size_t)row * M_PIX;

    for (int i = t; i < 65536; i += 1024) hist[i] = 0u;
    __syncthreads();

    // Phase 1: histogram of high 16 key bits
    for (int j = 0; j < 64; ++j) {
        int e = (j << 10) + t;
        if (e + 2048 < M_PIX) __builtin_prefetch(xr + e + 2048, 0, 1); // global_prefetch_b8
        atomicAdd(&hist[desc_key(xr[e]) >> 16], 1u);
    }
    __syncthreads();

    // Phase 2: exclusive scan of 65536 bins — per-thread chunk sums, then a
    // wave32 hierarchical shfl scan (replaces the serial thread-0 chain).
    uint32_t s = 0;
    const int b0 = t << 6;
    for (int j = 0; j < 64; ++j) s += hist[b0 + j];
    uint32_t inc = s;
    for (int d = 1; d < 32; d <<= 1) {
        int n = __shfl_up((int)inc, d);
        if (lane >= d) inc += (uint32_t)n;
    }
    if (lane == 31) wtot[w] = inc;
    __syncthreads();
    if (w == 0) {
        uint32_t wv = wtot[lane];
        uint32_t wi = wv;
        for (int d = 1; d < 32; d <<= 1) {
            int n = __shfl_up((int)wi, d);
            if (lane >= d) wi += (uint32_t)n;
        }
        wbase[lane] = wi - wv;           // exclusive wave base
    }
    __syncthreads();
    {
        uint32_t run = wbase[w] + (inc - s);   // exclusive prefix over chunk sums
        for (int j = 0; j < 64; ++j) { uint32_t v = hist[b0 + j]; hist[b0 + j] = run; run += v; }
    }
    __syncthreads();

    // Phase 3: scatter {lo16 | elem_idx16} records to provisional bucket slots
    for (int j = 0; j < 64; ++j) {
        int e = (j << 10) + t;
        uint32_t k    = desc_key(xr[e]);
        uint32_t slot = atomicAdd(&hist[k >> 16], 1u);
        recr[slot] = ((k & 0xFFFFu) << 16) | (uint32_t)e;
    }
    __threadfence();
    __syncthreads();
    // After phase 3: hist[h] == end of bucket h == start of bucket h+1.

    // Phase 4: exact rank within bucket + positional deposit into S
    for (int j = 0; j < 64; ++j) {
        int e = (j << 10) + t;
        float xv   = xr[e];
        float diff = xv - pr[e];
        float d    = diff * diff;
        uint32_t k     = desc_key(xv);
        uint32_t hi    = k >> 16;
        uint32_t start = hi ? hist[hi - 1] : 0u;
        uint32_t end   = hist[hi];
        uint32_t me    = ((k & 0xFFFFu) << 16) | (uint32_t)e;
        __builtin_prefetch(recr + start, 0, 0);
        uint32_t cnt = 0;
        for (uint32_t q = start; q < end; ++q) cnt += (recr[q] < me) ? 1u : 0u;
        atomicAdd(&S[start + cnt], d);
    }
}

// Single block, 32 waves. Wave w owns S[w*2048 .. w*2048+2048).
__global__ void finalize(const float* __restrict__ S, float* __restrict__ out, int nrows) {
    __shared__ float    waveSum[32], waveBase[32], candLoss[32];
    __shared__ uint32_t candIdx[32];
    const int t    = threadIdx.x;
    const int w    = t >> 5;
    const int lane = t & 31;
    const int off  = w << 11;

    // Stage 1: wave block-sum via f32 WMMA reduction (A = ones 16x4, exact).
    float total;
#if __has_builtin(__builtin_amdgcn_wmma_f32_16x16x4_f32)
    {
        v2f a; a.x = 1.f; a.y = 1.f;
        v8f c = {};
        for (int i = 0; i < 32; ++i) {
            const float* tp = S + off + (i << 6) + (lane << 1);
            v2f b; b.x = tp[0]; b.y = tp[1];
            // D[m][n] = sum_k B[k][n] + C[m][n]: accumulates column sums of each 4x16 tile
            c = __builtin_amdgcn_wmma_f32_16x16x4_f32(false, a, false, b,
                                                      (short)0, c, false, false);
        }
        float v = c[0];                  // identical rows; lanes hold column sums
        v += __shfl_xor(v, 1);
        v += __shfl_xor(v, 2);
        v += __shfl_xor(v, 4);
        v += __shfl_xor(v, 8);           // sum of 16 column-sum lanes = wave total
        total = v;
    }
#else
    {
        float v = 0.f;
        for (int i = lane; i < 2048; i += 32) v += S[off + i];
        v += __shfl_xor(v, 1); v += __shfl_xor(v, 2); v += __shfl_xor(v, 4);
        v += __shfl_xor(v, 8); v += __shfl_xor(v, 16);
        total = v;
    }
#endif
    if (lane == 0) { waveSum[w] = total; candIdx[w] = M_PIX + 1u; candLoss[w] = 0.f; }
    __syncthreads();
    if (t == 0) {
        float run = 0.f;
        for (int i = 0; i < 32; ++i) { float v = waveSum[i]; waveBase[i] = run; run += v; }
    }
    __syncthreads();

    // Stage 2: intra-wave running prefix scan + first-crossing search
    const float Nf = (float)nrows;
    float running = waveBase[w];
    for (int i = 0; i < 64; ++i) {
        float v = S[off + (i << 5) + lane];
        for (int d = 1; d < 32; d <<= 1) {          // wave32 inclusive scan
            float n = __shfl_up(v, d);
            if (lane >= d) v += n;
        }
        float csum = running + v;
        uint32_t idx = (uint32_t)(off + (i << 5) + lane + 1);
        float fi   = (float)idx;
        float loss = csum / (Nf * fi * fi);
        bool cond  = (idx >= AT_LEAST_I) && (loss >= THRESH);
        unsigned long long bal = __ballot(cond);
        if (bal) {
            int   fl   = __ffsll(bal) - 1;
            float lsel = __shfl(loss, fl);
            if (lane == 0) {
                candIdx[w]  = (uint32_t)(off + (i << 5) + fl + 1);
                candLoss[w] = lsel;
            }
            break;
        }
        running += __shfl(v, 31);
    }
    __syncthreads();

    if (t == 0) {
        uint32_t bi = M_PIX + 1u; float bl = 0.f;
        for (int i = 0; i < 32; ++i)
            if (candIdx[i] < bi) { bi = candIdx[i]; bl = candLoss[i]; }
        float loss; uint32_t iout;
        if (bi <= (uint32_t)M_PIX) { iout = bi; loss = bl; }
        else {
            iout = M_PIX;
            float grand = waveBase[31] + waveSum[31];
            float fm = (float)M_PIX;
            loss = grand / (Nf * fm * fm);
        }
        float thr = THRESH * ((iout == (uint32_t)M_PIX) ? 0.95f : 1.f)
                           * ((iout == AT_LEAST_I)      ? 1.05f : 1.f);
        out[0] = loss;
        out[1] = (float)iout;
        out[2] = thr;
    }
}

extern "C" void kernel_launch(void* const* d_in, const int* in_sizes, int n_in,
                              void* d_out, int out_size, void* d_ws, size_t ws_size,
                              hipStream_t stream) {
    const float* x = (const float*)d_in[0];
    const float* p = (const float*)d_in[1];
    float* out = (float*)d_out;
    const int total = in_sizes[0];
    const int nrows = total / M_PIX;          // 8*8*4 = 256

    float*    S   = (float*)d_ws;                                   // 256 KB
    uint32_t* rec = (uint32_t*)((char*)d_ws + (size_t)M_PIX * 4);   // nrows*256 KB

    const size_t ldsBytes = (65536 + 64) * sizeof(uint32_t);        // ~256 KB (CDNA5 LDS)
    (void)hipFuncSetAttribute(reinterpret_cast<const void*>(rank_deposit),
                              hipFuncAttributeMaxDynamicSharedMemorySize, (int)ldsBytes);

    zero_S<<<M_PIX / 1024, 1024, 0, stream>>>(S);
    rank_deposit<<<nrows, 1024, ldsBytes, stream>>>(x, p, rec, S);
    finalize<<<1, 1024, 0, stream>>>(S, out, nrows);
}